// LogisProcessor_6932077216291
// MI455X (gfx1250) — compile-verified
//
#include <hip/hip_runtime.h>
#include <hip/hip_bf16.h>
#include <stdint.h>

#define BATCH   256
#define VOCAB   50257
#define THREADS 256
#define CHUNK   ((VOCAB + THREADS - 1) / THREADS)   // 197 elems / thread for CDF pass
#define STAGE   2048                                 // floats staged per async round (8 KB)

// ---- gfx1250 async global->LDS path (probe-confirmed builtins, guarded) ----
#ifndef HAVE_ASYNC_LDS
#if defined(__has_builtin)
#if __has_builtin(__builtin_amdgcn_global_load_async_to_lds_b32) && \
    __has_builtin(__builtin_amdgcn_s_wait_asynccnt)
#define HAVE_ASYNC_LDS 1
#endif
#endif
#endif
#ifndef HAVE_ASYNC_LDS
#define HAVE_ASYNC_LDS 0
#endif

#ifndef HAVE_ASYNC_B128
#if defined(__has_builtin)
#if __has_builtin(__builtin_amdgcn_global_load_async_to_lds_b128)
#define HAVE_ASYNC_B128 1
#endif
#endif
#endif
#ifndef HAVE_ASYNC_B128
#define HAVE_ASYNC_B128 0
#endif

typedef int v4i __attribute__((vector_size(16)));
typedef __attribute__((address_space(1))) int* gas_i32_ptr;   // global, b32
typedef __attribute__((address_space(3))) int* las_i32_ptr;   // LDS,    b32
typedef __attribute__((address_space(1))) v4i* gas_v4i_ptr;   // global, b128
typedef __attribute__((address_space(3))) v4i* las_v4i_ptr;   // LDS,    b128

// Order-preserving float->u32 key: a > b  <=>  ordkey(a) > ordkey(b)
__device__ __forceinline__ unsigned ordkey(float f) {
  unsigned u = __float_as_uint(f);
  return u ^ ((u & 0x80000000u) ? 0xFFFFFFFFu : 0x80000000u);
}
__device__ __forceinline__ float keyval(unsigned k) {
  unsigned u = (k & 0x80000000u) ? (k ^ 0x80000000u) : ~k;
  return __uint_as_float(u);
}

// Deterministic per-row uniform in [0,1) (splitmix64 of (seed=42, row)).
__device__ __forceinline__ float row_uniform(int row) {
  uint64_t z = (uint64_t)(row + 1) * 0x9E3779B97F4A7C15ull + 42ull;
  z ^= z >> 30; z *= 0xBF58476D1CE4E5B9ull;
  z ^= z >> 27; z *= 0x94D049BB133111EBull;
  z ^= z >> 31;
  return (float)((z >> 40) * (1.0 / 16777216.0));   // 24-bit mantissa-safe
}

__global__ void __launch_bounds__(THREADS)
topk_sample_kernel(const float* __restrict__ logits,
                   const float* __restrict__ temp_p,
                   const int*   __restrict__ topk_p,
                   float*       __restrict__ out) {
  __shared__ __align__(16) float s_stage[STAGE];
  __shared__ unsigned s_hist[256];
  __shared__ float    s_csum[THREADS];     // chunk kept-exp sums (then effective sums)
  __shared__ unsigned s_cties[THREADS];    // chunk tie counts
  __shared__ float    s_cdf0[THREADS];     // exclusive scan of effective sums
  __shared__ unsigned s_tie0[THREADS];     // exclusive scan of tie counts
  __shared__ unsigned s_prefix, s_kk, s_maxkey;
  __shared__ float    s_total;
  __shared__ int      s_winner;

  const int row = blockIdx.x;
  const int tid = threadIdx.x;
  const float* __restrict__ g = logits + (size_t)row * VOCAB;
  const float invT = 1.0f / temp_p[0];
  unsigned K = (unsigned)topk_p[0];
  if (K < 1u) K = 1u;
  if (K > (unsigned)VOCAB) K = (unsigned)VOCAB;

  if (tid == 0) { s_prefix = 0u; s_kk = K; s_maxkey = 0u; s_winner = -1; }
  __syncthreads();

  // ---------------- exact radix select: find key of k-th largest ----------------
  for (int b = 3; b >= 0; --b) {
    s_hist[tid] = 0u;                           // THREADS == 256 bins
    __syncthreads();
    const unsigned prefix = s_prefix;
    const unsigned hmask  = (b == 3) ? 0u : (0xFFFFFFFFu << (8 * (b + 1)));

    if (b == 3) {
      // First pass streams from HBM through LDS with async B128 loads.
      unsigned lmax = 0u;
      // Per-row 16B alignment: VOCAB is odd, so odd rows start 4B-aligned.
      const int head = (int)(((16u - ((unsigned)(uintptr_t)g & 15u)) & 15u) >> 2);
      if (tid < head) {                         // head < 4 elements
        const unsigned k = ordkey(g[tid]);
        lmax = max(lmax, k);
        atomicAdd(&s_hist[k >> 24], 1u);
      }
      int base = head;
#if HAVE_ASYNC_LDS && HAVE_ASYNC_B128
      for (; base + STAGE <= VOCAB; base += STAGE) {
        // 2048 floats = 512 B128 loads = 2 per thread; 16B-aligned on both ends.
        #pragma unroll
        for (int t = 0; t < STAGE / (4 * THREADS); ++t) {
          const int e = (tid + t * THREADS) * 4;          // offset within chunk
          __builtin_amdgcn_global_load_async_to_lds_b128(
              (gas_v4i_ptr)(g + base + e),
              (las_v4i_ptr)(&s_stage[e]), 0, 0);
        }
        __builtin_amdgcn_s_wait_asynccnt(0);              // this wave's loads done
        __syncthreads();                                  // => all waves' loads done
        for (int i = tid * 4; i < STAGE; i += THREADS * 4) {
          const float4 v = *(const float4*)&s_stage[i];   // ds_load_b128
          unsigned k0 = ordkey(v.x), k1 = ordkey(v.y);
          unsigned k2 = ordkey(v.z), k3 = ordkey(v.w);
          lmax = max(max(lmax, max(k0, k1)), max(k2, k3));
          atomicAdd(&s_hist[k0 >> 24], 1u);
          atomicAdd(&s_hist[k1 >> 24], 1u);
          atomicAdd(&s_hist[k2 >> 24], 1u);
          atomicAdd(&s_hist[k3 >> 24], 1u);
        }
        __syncthreads();                                  // buffer reuse
      }
#endif
      // Tail (and the whole row if B128 path unavailable): B32 async staging.
      for (; base < VOCAB; base += STAGE) {
        const int n = min(STAGE, VOCAB - base);
#if HAVE_ASYNC_LDS
        for (int i = tid; i < n; i += THREADS)
          __builtin_amdgcn_global_load_async_to_lds_b32(
              (gas_i32_ptr)(g + base + i),
              (las_i32_ptr)(&s_stage[i]), 0, 0);
        __builtin_amdgcn_s_wait_asynccnt(0);
#else
        for (int i = tid; i < n; i += THREADS) s_stage[i] = g[base + i];
#endif
        __syncthreads();
        for (int i = tid; i < n; i += THREADS) {
          const unsigned k = ordkey(s_stage[i]);
          lmax = max(lmax, k);
          atomicAdd(&s_hist[k >> 24], 1u);
        }
        __syncthreads();
      }
      atomicMax(&s_maxkey, lmax);
    } else {
      // Refinement passes: 51.5 MB fits in 192 MB L2 -> these are L2-bandwidth.
      for (int i = tid; i < VOCAB; i += THREADS) {
        if (i + 4 * THREADS < VOCAB) __builtin_prefetch(g + i + 4 * THREADS, 0, 1);
        const unsigned k = ordkey(g[i]);
        if ((k & hmask) == (prefix & hmask))
          atomicAdd(&s_hist[(k >> (8 * b)) & 255u], 1u);
      }
    }
    __syncthreads();
    if (tid == 0) {
      unsigned kk = s_kk;
      int d = 0;
      for (int dd = 255; dd >= 0; --dd) {
        const unsigned c = s_hist[dd];
        if (kk <= c) { d = dd; break; }
        kk -= c;
      }
      s_prefix = prefix | ((unsigned)d << (8 * b));
      s_kk = kk;
    }
    __syncthreads();
  }

  const unsigned T     = s_prefix;   // key of the k-th largest element
  const unsigned tkeep = s_kk;       // # ties at T to keep (lowest indices first)
  const float M  = keyval(s_maxkey); // row max (top-1 is inside top-k)
  const float eT = __expf((keyval(T) - M) * invT);

  // ---------------- per-chunk kept-exp sums + tie counts (contiguous chunks) ----
  const int beg = tid * CHUNK;
  const int end = min(VOCAB, beg + CHUNK);
  float csum = 0.0f; unsigned cties = 0u;
  for (int i = beg; i < end; ++i) {
    const float x = g[i];
    const unsigned k = ordkey(x);
    if (k > T)       csum += __expf((x - M) * invT);
    else if (k == T) cties++;
  }
  s_csum[tid] = csum; s_cties[tid] = cties;
  __syncthreads();

  // Serial exclusive scans (256 entries, trivial cost vs 50K-elem passes).
  if (tid == 0) {
    unsigned tacc = 0u;
    for (int i = 0; i < THREADS; ++i) { s_tie0[i] = tacc; tacc += s_cties[i]; }
    float acc = 0.0f;
    for (int i = 0; i < THREADS; ++i) {
      const unsigned tb = s_tie0[i];
      unsigned keep = 0u;
      if (tkeep > tb) {
        const unsigned rem = tkeep - tb;
        keep = (rem < s_cties[i]) ? rem : s_cties[i];
      }
      const float eff = s_csum[i] + (float)keep * eT;
      s_cdf0[i] = acc;
      s_csum[i] = eff;                 // overwrite with effective (tie-limited) sum
      acc += eff;
    }
    s_total = acc;
  }
  __syncthreads();

  // ---------------- inverse-CDF sample ----------------
  const float S      = s_total;
  const float target = row_uniform(row) * S;
  const float lo  = s_cdf0[tid];
  const float eff = s_csum[tid];
  if (eff > 0.0f && target >= lo && target < lo + eff) atomicMax(&s_winner, tid);
  __syncthreads();
  if (tid == 0 && s_winner < 0) {      // fp edge: target landed past the total
    for (int i = THREADS - 1; i >= 0; --i)
      if (s_csum[i] > 0.0f) { s_winner = i; break; }
  }
  __syncthreads();

  if (tid == s_winner) {
    float acc = lo;
    unsigned trank = s_tie0[tid];
    int pick = -1;
    for (int i = beg; i < end; ++i) {
      const float x = g[i];
      const unsigned k = ordkey(x);
      float p = 0.0f;
      if (k > T)        p = __expf((x - M) * invT);
      else if (k == T) { if (trank < tkeep) p = eT; trank++; }
      if (p > 0.0f) { pick = i; acc += p; if (acc > target) break; }
    }
    if (pick < 0) pick = beg;
    out[row] = (float)pick;
  }
}

extern "C" void kernel_launch(void* const* d_in, const int* in_sizes, int n_in,
                              void* d_out, int out_size, void* d_ws, size_t ws_size,
                              hipStream_t stream) {
  (void)in_sizes; (void)n_in; (void)d_ws; (void)ws_size; (void)out_size;
  const float* logits = (const float*)d_in[0];
  const float* temp   = (const float*)d_in[1];
  const int*   topk   = (const int*)d_in[2];
  float*       out    = (float*)d_out;
  topk_sample_kernel<<<BATCH, THREADS, 0, stream>>>(logits, temp, topk, out);
}